// _RoIOffsetPooling_78907139162171
// MI455X (gfx1250) — compile-verified
//
#include <hip/hip_runtime.h>
#include <hip/hip_bf16.h>
#include <math.h>

// Problem constants from the reference
#define NB    4
#define CCH   256
#define HH    38
#define WW    38
#define RR    128
#define PH    7
#define PW    7
#define SCALE 0.0625f
#define GAMMA 0.1f

#define MROWS (RR * CCH)      // 32768
#define KDIM  (PH * PW)       // 49
#define NDIM  (2 * PH * PW)   // 98

typedef __attribute__((ext_vector_type(2))) float v2f;
typedef __attribute__((ext_vector_type(8))) float v8f;

// ---------------------------------------------------------------------------
// Kernel 1: RoI max pooling -> pooled (MROWS x KDIM), row m = r*C + c,
// col = ph*PW + pw  (matches reference reshape(R*C, PH*PW))
// ---------------------------------------------------------------------------
__global__ __launch_bounds__(256) void roi_pool_kernel(
    const float* __restrict__ feat, const float* __restrict__ rois,
    float* __restrict__ pooled)
{
    int t = blockIdx.x * blockDim.x + threadIdx.x;
    if (t >= MROWS * KDIM) return;
    int pw = t % PW;
    int ph = (t / PW) % PH;
    int c  = (t / KDIM) % CCH;
    int r  = t / (KDIM * CCH);

    const float* roi = rois + r * 5;
    int b  = (int)roi[0];
    int x1 = (int)rintf(roi[1] * SCALE);
    int y1 = (int)rintf(roi[2] * SCALE);
    int x2 = (int)rintf(roi[3] * SCALE);
    int y2 = (int)rintf(roi[4] * SCALE);
    float bin_h = fmaxf((float)(y2 - y1 + 1), 1.0f) / (float)PH;
    float bin_w = fmaxf((float)(x2 - x1 + 1), 1.0f) / (float)PW;

    int hs = min(max(y1 + (int)floorf((float)ph * bin_h), 0), HH);
    int he = min(max(y1 + (int)ceilf((float)(ph + 1) * bin_h), 0), HH);
    int ws = min(max(x1 + (int)floorf((float)pw * bin_w), 0), WW);
    int we = min(max(x1 + (int)ceilf((float)(pw + 1) * bin_w), 0), WW);

    const float* f = feat + ((size_t)b * CCH + c) * (HH * WW);
    float m = -INFINITY;
    #pragma unroll
    for (int kh = 0; kh < 7; ++kh) {
        int h = hs + kh;
        if (h >= he) continue;
        #pragma unroll
        for (int kw = 0; kw < 7; ++kw) {
            int w = ws + kw;
            if (w >= we) continue;
            m = fmaxf(m, f[h * WW + w]);
        }
    }
    pooled[t] = isinf(m) ? 0.0f : m;
}

// ---------------------------------------------------------------------------
// Kernel 2: offsets = (pooled @ fc_w^T + fc_b) * GAMMA  via V_WMMA_F32_16X16X4_F32
// Each wave: one 16-row M tile, all 7 N tiles (N=98 padded to 112), K=49 -> 13
// k-steps of 4.  A fragment reused across the 7 N tiles -> 91 WMMAs per wave.
// ---------------------------------------------------------------------------
__global__ __launch_bounds__(256) void gemm_offset_kernel(
    const float* __restrict__ pooled, const float* __restrict__ fcw,
    const float* __restrict__ fcb, float* __restrict__ offs)
{
    const int lane   = threadIdx.x & 31;
    const int wave   = blockIdx.x * (blockDim.x >> 5) + (threadIdx.x >> 5);
    const int m_base = wave * 16;
    const int half   = lane >> 4;   // 0: K pair {k0,k0+1}, 1: {k0+2,k0+3}
    const int l15    = lane & 15;   // A: row-in-tile, B/D: col-in-tile

    v8f acc[7] = {};

    const int arow = m_base + l15;
    for (int ks = 0; ks < 13; ++ks) {
        const int k0 = ks * 4 + half * 2;
        v2f a;
        a.x = (k0     < KDIM) ? pooled[arow * KDIM + k0]     : 0.0f;
        a.y = (k0 + 1 < KDIM) ? pooled[arow * KDIM + k0 + 1] : 0.0f;
        #pragma unroll
        for (int nt = 0; nt < 7; ++nt) {
            const int n = nt * 16 + l15;
            const bool nok = (n < NDIM);
            v2f bf;
            bf.x = (nok && k0     < KDIM) ? fcw[n * KDIM + k0]     : 0.0f;
            bf.y = (nok && k0 + 1 < KDIM) ? fcw[n * KDIM + k0 + 1] : 0.0f;
            // D = A(16x4) * B(4x16) + C, full f32 precision
            acc[nt] = __builtin_amdgcn_wmma_f32_16x16x4_f32(
                false, a, false, bf, (short)0, acc[nt], false, false);
        }
    }

    #pragma unroll
    for (int nt = 0; nt < 7; ++nt) {
        const int n = nt * 16 + l15;
        if (n < NDIM) {
            const float bias = fcb[n];
            #pragma unroll
            for (int v = 0; v < 8; ++v) {
                const int m = m_base + v + half * 8;   // C/D layout: VGPR v -> M
                offs[(size_t)m * NDIM + n] = (acc[nt][v] + bias) * GAMMA;
            }
        }
    }
}

// ---------------------------------------------------------------------------
// Kernel 3: deformable bilinear sampling
// ---------------------------------------------------------------------------
__global__ __launch_bounds__(256) void deform_pool_kernel(
    const float* __restrict__ feat, const float* __restrict__ rois,
    const float* __restrict__ offs, float* __restrict__ out)
{
    int t = blockIdx.x * blockDim.x + threadIdx.x;
    if (t >= MROWS * KDIM) return;
    int pw = t % PW;
    int ph = (t / PW) % PH;
    int c  = (t / KDIM) % CCH;
    int r  = t / (KDIM * CCH);

    const float* roi = rois + r * 5;
    int   b  = (int)roi[0];
    float x1 = rintf(roi[1] * SCALE);
    float y1 = rintf(roi[2] * SCALE);
    float x2 = rintf(roi[3] * SCALE);
    float y2 = rintf(roi[4] * SCALE);
    float bin_h = fmaxf(y2 - y1 + 1.0f, 1.0f) / (float)PH;
    float bin_w = fmaxf(x2 - x1 + 1.0f, 1.0f) / (float)PW;
    float cy = y1 + ((float)ph + 0.5f) * bin_h;
    float cx = x1 + ((float)pw + 0.5f) * bin_w;

    // offsets row (r*C+c), cols (ph*PW+pw)*2 + {0:y, 1:x}
    const float* orow = offs + (size_t)(r * CCH + c) * NDIM + (ph * PW + pw) * 2;
    float ys = cy + orow[0];
    float xs = cx + orow[1];

    float y0f = floorf(ys), x0f = floorf(xs);
    float wy = ys - y0f, wx = xs - x0f;
    int y0 = (int)y0f, x0 = (int)x0f;

    const float* f = feat + ((size_t)b * CCH + c) * (HH * WW);
    auto gather = [&](int yi, int xi) -> float {
        bool inb = (yi >= 0) && (yi < HH) && (xi >= 0) && (xi < WW);
        int yc = min(max(yi, 0), HH - 1);
        int xc = min(max(xi, 0), WW - 1);
        return inb ? f[yc * WW + xc] : 0.0f;
    };

    out[t] = gather(y0, x0)         * (1.0f - wy) * (1.0f - wx)
           + gather(y0, x0 + 1)     * (1.0f - wy) * wx
           + gather(y0 + 1, x0)     * wy          * (1.0f - wx)
           + gather(y0 + 1, x0 + 1) * wy          * wx;
}

// ---------------------------------------------------------------------------
extern "C" void kernel_launch(void* const* d_in, const int* in_sizes, int n_in,
                              void* d_out, int out_size, void* d_ws, size_t ws_size,
                              hipStream_t stream) {
    const float* features = (const float*)d_in[0];  // (4,256,38,38)
    const float* rois     = (const float*)d_in[1];  // (128,5)
    const float* fcw      = (const float*)d_in[2];  // (98,49)
    const float* fcb      = (const float*)d_in[3];  // (98,)
    float* out = (float*)d_out;                     // (128,256,7,7)

    // workspace: pooled (MROWS*KDIM) then offsets (MROWS*NDIM) -> ~19.3 MB
    float* pooled = (float*)d_ws;
    float* offs   = pooled + (size_t)MROWS * KDIM;

    const int total = MROWS * KDIM;  // 1,605,632

    roi_pool_kernel<<<(total + 255) / 256, 256, 0, stream>>>(features, rois, pooled);

    // 2048 M-tiles of 16 rows, 8 waves per 256-thread block -> 256 blocks
    gemm_offset_kernel<<<MROWS / 16 / 8, 256, 0, stream>>>(pooled, fcw, fcb, offs);

    deform_pool_kernel<<<(total + 255) / 256, 256, 0, stream>>>(features, rois, offs, out);
}